// StatePredictor_65438121722448
// MI455X (gfx1250) — compile-verified
//
#include <hip/hip_runtime.h>

// CDNA5 / gfx1250, wave32. f32 WMMA 16x16x4 for both the Gram-trick KNN
// distance tiles and the 3-layer MLP (keeps reference fp32 numerics).
// 4 waves per block cooperate on ONE 16-query tile: each wave scans a strided
// quarter of the 512 candidate tiles (occupancy 4x, per-wave latency /4),
// then an 8-way merge of per-lane top-11 lists produces the final neighbors.
// The candidate stream is software double-buffered; the last iteration is
// peeled so the steady-state prefetch is a constant-offset b96 load.
typedef __attribute__((ext_vector_type(2))) float v2f;
typedef __attribute__((ext_vector_type(8))) float v8f;

#define NPTS   8192
#define BATCH  4
#define NSEL   11          // K+1 neighbors (incl. self)
#define NTILES (NPTS / 16) // 512 candidate tiles
#define WAVES  4           // waves per block, all on the same query tile

__global__ __launch_bounds__(WAVES * 32)
void state_predictor_kernel(const float* __restrict__ pos,   // [B,N,3]
                            const float* __restrict__ vel,   // [B,N,3]
                            const float* __restrict__ box,   // [B]
                            const float* __restrict__ W1,    // [64,32]
                            const float* __restrict__ b1,    // [32]
                            const float* __restrict__ W2,    // [32,16]
                            const float* __restrict__ b2,    // [16]
                            const float* __restrict__ W3,    // [16,6]
                            const float* __restrict__ b3,    // [6]
                            float* __restrict__ out)         // [B,N,6]
{
    __shared__ float sD[WAVES][32][NSEL];   // per-wave per-lane top-11 dists
    __shared__ int   sI[WAVES][32][NSEL];   // per-wave per-lane top-11 indices
    __shared__ int   sNN[16][NSEL];         // merged neighbor indices
    __shared__ float sQP[16][3];            // query positions
    __shared__ float sF[16][64];            // feature matrix
    __shared__ float sH1[16][32];           // layer-1 activations
    __shared__ float sH2[16][16];           // layer-2 activations

    const int tid   = threadIdx.x;
    const int w     = tid >> 5;
    const int lane  = tid & 31;
    const int half  = lane >> 4;   // 0 = lanes 0-15, 1 = lanes 16-31
    const int l16   = lane & 15;
    const int b     = blockIdx.x >> 9;            // / NTILES
    const int qt    = blockIdx.x & (NTILES - 1);
    const int qbase = qt * 16;

    const float* posb = pos + (size_t)b * NPTS * 3;
    const float* velb = vel + (size_t)b * NPTS * 3;

    // ------- query operand: B matrix column n holds (-2qx,-2qy,-2qz,1) -----
    const int q = qbase + l16;
    const float qx = posb[q * 3 + 0];
    const float qy = posb[q * 3 + 1];
    const float qz = posb[q * 3 + 2];
    const float qsq = qx * qx + qy * qy + qz * qz;
    v2f Bq;                                        // branchless per-half pick
    Bq.x = half ? (-2.0f * qz) : (-2.0f * qx);     // K=2 | K=0
    Bq.y = half ? 1.0f         : (-2.0f * qy);     // K=3 | K=1
    if (w == 0 && half == 0) {
        sQP[l16][0] = qx; sQP[l16][1] = qy; sQP[l16][2] = qz;
    }

    // ------- per-lane running top-11 (register-resident, sorted) -----------
    float bd[NSEL]; int bi[NSEL];
#pragma unroll
    for (int j = 0; j < NSEL; ++j) { bd[j] = 3.4e38f; bi[j] = 0; }

    // one candidate tile: |c|^2 via VALU, d2 tile via WMMA, guarded inserts
    auto process_tile = [&](int ct, float cx, float cy, float cz) {
        const float csq = cx * cx + cy * cy + cz * cz;
        v2f Ac;                                    // A row m = (cx,cy,cz,|c|^2)
        Ac.x = half ? cz  : cx;                    // K=2 | K=0 (branchless)
        Ac.y = half ? csq : cy;                    // K=3 | K=1
        v8f acc = {};
        acc = __builtin_amdgcn_wmma_f32_16x16x4_f32(false, Ac, false, Bq,
                                                    (short)0, acc, false, false);
        const int m0 = ct * 16;
        // lane holds d2 for candidates m = m0 + v + 8*half, query n = l16
#pragma unroll
        for (int v = 0; v < 8; ++v) {
            const float d  = acc[v] + qsq;         // |c|^2 - 2 c.q + |q|^2
            const int  idx = m0 + v + 8 * half;
            if (d < bd[NSEL - 1]) {                // amortized-rare, EXEC-masked
                float nd = d; int ni = idx;        // branchless ripple insert
#pragma unroll
                for (int j = 0; j < NSEL; ++j) {
                    const bool cl = nd < bd[j];
                    const float oj = bd[j]; const int xj = bi[j];
                    bd[j] = cl ? nd : oj;
                    bi[j] = cl ? ni : xj;
                    nd    = cl ? oj : nd;
                    ni    = cl ? xj : ni;
                }
            }
        }
    };

    // ------- KNN: wave scans candidate tiles w, w+4, ..., double-buffered --
    {
        constexpr int STRIDE_F = WAVES * 16 * 3;   // floats per wave step
        int ct = w;
        const float* cp = posb + (size_t)(ct * 16 + l16) * 3;
        float cx = cp[0], cy = cp[1], cz = cp[2];
        for (; ct + WAVES < NTILES; ct += WAVES) {
            const float nx = cp[STRIDE_F + 0];     // constant-offset prefetch
            const float ny = cp[STRIDE_F + 1];     // stays in flight behind
            const float nz = cp[STRIDE_F + 2];     // the WMMA + inserts
            process_tile(ct, cx, cy, cz);
            cx = nx; cy = ny; cz = nz;
            cp += STRIDE_F;
        }
        process_tile(ct, cx, cy, cz);              // peeled final tile
    }

    // ------- publish per-lane lists, then 8-way merge (wave 0, lanes 0-15) -
#pragma unroll
    for (int j = 0; j < NSEL; ++j) { sD[w][lane][j] = bd[j]; sI[w][lane][j] = bi[j]; }
    __syncthreads();
    if (w == 0 && half == 0) {
        int p[2 * WAVES];
#pragma unroll
        for (int l = 0; l < 2 * WAVES; ++l) p[l] = 0;
#pragma unroll
        for (int j = 0; j < NSEL; ++j) {
            float best = 3.4e38f; int bix = 0x7fffffff; int bl = 0;
#pragma unroll
            for (int l = 0; l < 2 * WAVES; ++l) {       // 8 sorted lists
                const int wv = l >> 1;
                const int ln = l16 + 16 * (l & 1);
                const float dv = sD[wv][ln][p[l]];
                const int   iv = sI[wv][ln][p[l]];
                const bool better = (dv < best) || ((dv == best) && (iv < bix));
                best = better ? dv : best;
                bix  = better ? iv : bix;
                bl   = better ? l  : bl;
            }
            sNN[l16][j] = bix;
#pragma unroll
            for (int l = 0; l < 2 * WAVES; ++l) p[l] += (l == bl) ? 1 : 0;
        }
    }
    __syncthreads();

    // ------- feature assembly: [velnn 33 | posoff 30 | boxdim 1] -----------
    for (int t = tid; t < 16 * NSEL; t += WAVES * 32) {  // 176 gather tasks
        const int r = t / NSEL, j = t % NSEL;
        const int idx = sNN[r][j];
        sF[r][3 * j + 0] = velb[idx * 3 + 0];
        sF[r][3 * j + 1] = velb[idx * 3 + 1];
        sF[r][3 * j + 2] = velb[idx * 3 + 2];
        if (j >= 1) {                      // drop self for position offsets
            const int o = 33 + 3 * (j - 1);
            sF[r][o + 0] = posb[idx * 3 + 0] - sQP[r][0];
            sF[r][o + 1] = posb[idx * 3 + 1] - sQP[r][1];
            sF[r][o + 2] = posb[idx * 3 + 2] - sQP[r][2];
        }
    }
    if (tid < 16) sF[tid][63] = box[b];
    __syncthreads();

    // ------- MLP on wave 0 only (44 WMMAs; negligible vs. the KNN scan) ----
    if (w == 0) {
        // layer 1: [16x64] @ [64x32], 16 K-steps x 2 N-tiles
        v8f c0 = {}, c1 = {};
#pragma unroll
        for (int s = 0; s < 16; ++s) {
            const int k0 = 4 * s + 2 * half;       // K slice per lane-half
            v2f A;  A.x  = sF[l16][k0];      A.y  = sF[l16][k0 + 1];
            v2f B0; B0.x = W1[(k0 + 0) * 32 + l16];       B0.y = W1[(k0 + 1) * 32 + l16];
            v2f B1; B1.x = W1[(k0 + 0) * 32 + 16 + l16];  B1.y = W1[(k0 + 1) * 32 + 16 + l16];
            c0 = __builtin_amdgcn_wmma_f32_16x16x4_f32(false, A, false, B0, (short)0, c0, false, false);
            c1 = __builtin_amdgcn_wmma_f32_16x16x4_f32(false, A, false, B1, (short)0, c1, false, false);
        }
        const float g0 = b1[l16], g1 = b1[16 + l16];
#pragma unroll
        for (int v = 0; v < 8; ++v) {
            const int row = v + 8 * half;
            float h0 = c0[v] + g0; h0 = h0 > 0.f ? h0 : 0.f;
            float h1 = c1[v] + g1; h1 = h1 > 0.f ? h1 : 0.f;
            sH1[row][l16]      = h0;
            sH1[row][16 + l16] = h1;
        }
    }
    __syncthreads();

    if (w == 0) {
        // layer 2: [16x32] @ [32x16], 8 K-steps
        v8f c2 = {};
#pragma unroll
        for (int s = 0; s < 8; ++s) {
            const int k0 = 4 * s + 2 * half;
            v2f A;  A.x  = sH1[l16][k0];  A.y  = sH1[l16][k0 + 1];
            v2f Bm; Bm.x = W2[(k0 + 0) * 16 + l16]; Bm.y = W2[(k0 + 1) * 16 + l16];
            c2 = __builtin_amdgcn_wmma_f32_16x16x4_f32(false, A, false, Bm, (short)0, c2, false, false);
        }
        const float g = b2[l16];
#pragma unroll
        for (int v = 0; v < 8; ++v) {
            const int row = v + 8 * half;
            float h = c2[v] + g; h = h > 0.f ? h : 0.f;
            sH2[row][l16] = h;
        }
    }
    __syncthreads();

    if (w == 0) {
        // layer 3: [16x16] @ [16x6 pad 16], 4 K-steps
        v8f c3 = {};
#pragma unroll
        for (int s = 0; s < 4; ++s) {
            const int k0 = 4 * s + 2 * half;
            v2f A;  A.x = sH2[l16][k0]; A.y = sH2[l16][k0 + 1];
            const bool valid = l16 < 6;            // branchless pad to N=16
            v2f Bm;
            Bm.x = valid ? W3[(k0 + 0) * 6 + l16] : 0.f;
            Bm.y = valid ? W3[(k0 + 1) * 6 + l16] : 0.f;
            c3 = __builtin_amdgcn_wmma_f32_16x16x4_f32(false, A, false, Bm, (short)0, c3, false, false);
        }
        if (l16 < 6) {
            const float g = b3[l16];
#pragma unroll
            for (int v = 0; v < 8; ++v) {
                const int row = v + 8 * half;
                float val = c3[v] + g;
                if (l16 < 3) val += sQP[row][l16]; // pred[:3] + position
                out[((size_t)b * NPTS + qbase + row) * 6 + l16] = val;
            }
        }
    }
}

extern "C" void kernel_launch(void* const* d_in, const int* in_sizes, int n_in,
                              void* d_out, int out_size, void* d_ws, size_t ws_size,
                              hipStream_t stream) {
    (void)in_sizes; (void)n_in; (void)d_ws; (void)ws_size; (void)out_size;
    const float* pos = (const float*)d_in[0];
    const float* vel = (const float*)d_in[1];
    const float* box = (const float*)d_in[2];
    const float* W1  = (const float*)d_in[3];
    const float* b1  = (const float*)d_in[4];
    const float* W2  = (const float*)d_in[5];
    const float* b2  = (const float*)d_in[6];
    const float* W3  = (const float*)d_in[7];
    const float* b3  = (const float*)d_in[8];
    float* out = (float*)d_out;

    const int blocks = BATCH * NTILES;   // 2048 blocks x 128 thr = 8192 waves
    state_predictor_kernel<<<blocks, WAVES * 32, 0, stream>>>(
        pos, vel, box, W1, b1, W2, b2, W3, b3, out);
}